// GraphConvBlock_83597243449881
// MI455X (gfx1250) — compile-verified
//
#include <hip/hip_runtime.h>

typedef __attribute__((ext_vector_type(2))) float v2f;
typedef __attribute__((ext_vector_type(4))) float v4f;
typedef __attribute__((ext_vector_type(8))) float v8f;

// ---------------------------------------------------------------------------
// Layout transforms
// ---------------------------------------------------------------------------

// NCHW -> NHWC (output-coalesced)
__global__ __launch_bounds__(256) void k_nchw_to_nhwc(
    const float* __restrict__ src, float* __restrict__ dst,
    int N, int C, int H, int W) {
  int i = blockIdx.x * blockDim.x + threadIdx.x;
  int total = N * C * H * W;
  if (i >= total) return;
  int c = i % C;
  int x = (i / C) % W;
  int y = (i / (C * W)) % H;
  int n = i / (C * W * H);
  dst[i] = src[((n * C + c) * H + y) * W + x];
}

// [L][CO][CI][3][3] (OIHW) -> [L][9 taps][CO][CI]  (ci innermost => b128 A loads)
__global__ __launch_bounds__(256) void k_weight_transform(
    const float* __restrict__ src, float* __restrict__ dst,
    int L, int CO, int CI) {
  int i = blockIdx.x * blockDim.x + threadIdx.x;
  int total = L * 9 * CO * CI;
  if (i >= total) return;
  int ci  = i % CI;
  int co  = (i / CI) % CO;
  int tap = (i / (CI * CO)) % 9;
  int l   = i / (CI * CO * 9);
  dst[i] = src[((l * CO + co) * CI + ci) * 9 + tap];
}

// ---------------------------------------------------------------------------
// 3x3 SAME conv as implicit GEMM on V_WMMA_F32_16X16X4_F32.
//   inA (+ optional inB, summed on load): NHWC [8,64,64,CIN]
//   wt: [9][128][CIN]   bias: [128]   out: NHWC or NCHW [8,128,64,64]
// One wave => 16(co) x 64(w) strip of one (n,h) row. 8 waves/block = all co.
// K-chunking: 8 channels per iteration -> one b128 A load + one b128 B load
// per tile feed two WMMAs (channel permutation applied identically to A & B).
// ---------------------------------------------------------------------------
template <int CIN, bool HAS2, bool NCHW_OUT>
__global__ __launch_bounds__(256) void k_conv3x3(
    const float* __restrict__ inA, const float* __restrict__ inB,
    const float* __restrict__ wt, const float* __restrict__ bias,
    float biasMult, float* __restrict__ out) {
  constexpr int H = 64, W = 64, CO = 128;
  const int lane = threadIdx.x & 31;
  const int wv   = threadIdx.x >> 5;   // co tile 0..7
  const int lm   = lane & 15;          // M (A/co) or N (B/w) index within tile
  const int lg   = lane >> 4;          // lane half-group
  const int row  = blockIdx.x;         // n*H + h
  const int n    = row >> 6;
  const int h    = row & 63;
  const int co0  = wv * 16;

  v8f acc[4];
#pragma unroll
  for (int t = 0; t < 4; ++t) acc[t] = (v8f){0, 0, 0, 0, 0, 0, 0, 0};

  for (int kh = 0; kh < 3; ++kh) {
    const int y = h + kh - 1;
    if ((unsigned)y >= (unsigned)H) continue;  // uniform: EXEC stays all-1s
    const float* rA = inA + (size_t)((n * H + y) * W) * CIN;
    const float* rB = HAS2 ? (inB + (size_t)((n * H + y) * W) * CIN) : nullptr;
#pragma unroll
    for (int kw = 0; kw < 3; ++kw) {
      const int tap = kh * 3 + kw;
      const float* wp = wt + (size_t)(tap * CO + co0 + lm) * CIN + 4 * lg;
      const float* pA[4];
      const float* pB[4];
      float msk[4];
#pragma unroll
      for (int t = 0; t < 4; ++t) {
        int x = 16 * t + lm + kw - 1;
        bool v = (unsigned)x < (unsigned)W;
        msk[t] = v ? 1.0f : 0.0f;        // per-lane cndmask, no divergence
        int xc = v ? x : 0;
        pA[t] = rA + (size_t)xc * CIN + 4 * lg;
        if (HAS2) pB[t] = rB + (size_t)xc * CIN + 4 * lg;
      }
#pragma unroll 4
      for (int ci = 0; ci < CIN; ci += 8) {
        v4f aw = *(const v4f*)(wp + ci);           // weights: 4 consecutive ci
        v2f a01 = {aw.x, aw.y};
        v2f a23 = {aw.z, aw.w};
#pragma unroll
        for (int t = 0; t < 4; ++t) {
          v4f b = *(const v4f*)(pA[t] + ci);       // acts: 4 consecutive ci
          if (HAS2) b = b + *(const v4f*)(pB[t] + ci);  // fused child sum
          b = b * msk[t];                          // zero OOB columns
          v2f b01 = {b.x, b.y};
          v2f b23 = {b.z, b.w};
          acc[t] = __builtin_amdgcn_wmma_f32_16x16x4_f32(
              false, a01, false, b01, (short)0, acc[t], false, false);
          acc[t] = __builtin_amdgcn_wmma_f32_16x16x4_f32(
              false, a23, false, b23, (short)0, acc[t], false, false);
        }
      }
    }
  }

  // Bias + store. C/D layout: vgpr r, lane -> (co = co0 + r + 8*lg, w = 16t+lm)
  v4f blo = *(const v4f*)(bias + co0 + 8 * lg);
  v4f bhi = *(const v4f*)(bias + co0 + 8 * lg + 4);
#pragma unroll
  for (int r = 0; r < 8; ++r) {
    const int co = co0 + 8 * lg + r;
    const float bv = biasMult * ((r < 4) ? blo[r] : bhi[r - 4]);
#pragma unroll
    for (int t = 0; t < 4; ++t) {
      const int x = 16 * t + lm;
      const float v = acc[t][r] + bv;
      if (NCHW_OUT)
        out[((size_t)(n * CO + co) * H + h) * W + x] = v;
      else
        out[((size_t)((n * H + h) * W + x)) * CO + co] = v;
    }
  }
}

// ---------------------------------------------------------------------------
// Host: 3 transform launches + 7 fused convs (12 reference convs folded to 7
// via linearity: conv(a,W)+conv(b,W) = conv(a+b,W), bias * #children).
// ---------------------------------------------------------------------------
extern "C" void kernel_launch(void* const* d_in, const int* in_sizes, int n_in,
                              void* d_out, int out_size, void* d_ws,
                              size_t ws_size, hipStream_t stream) {
  (void)in_sizes; (void)n_in; (void)out_size; (void)ws_size;
  const float* x  = (const float*)d_in[0];  // [8,64,64,64]
  const float* W0 = (const float*)d_in[1];  // [128,64,3,3]
  const float* b0 = (const float*)d_in[2];  // [128]
  const float* Wn = (const float*)d_in[3];  // [6,128,128,3,3]
  const float* bn = (const float*)d_in[4];  // [6,128]
  float* out = (float*)d_out;               // [8,128,64,64] NCHW
  float* ws  = (float*)d_ws;

  constexpr size_t XT  = 8ull * 64 * 64 * 64;     // x in NHWC
  constexpr size_t W0T = 9ull * 128 * 64;         // stem weights [tap][co][ci]
  constexpr size_t WNT = 6ull * 9 * 128 * 128;    // node weights
  constexpr size_t ACT = 8ull * 64 * 64 * 128;    // one NHWC activation
  constexpr size_t WL  = 9ull * 128 * 128;        // weights per node layer

  float* xt   = ws;
  float* w0t  = xt + XT;
  float* wnt  = w0t + W0T;
  float* buf0 = wnt + WNT;                        // 3 rotating activation bufs
  float* buf1 = buf0 + ACT;
  float* buf2 = buf1 + ACT;

  k_nchw_to_nhwc<<<(XT + 255) / 256, 256, 0, stream>>>(x, xt, 8, 64, 64, 64);
  k_weight_transform<<<(W0T + 255) / 256, 256, 0, stream>>>(W0, w0t, 1, 128, 64);
  k_weight_transform<<<(WNT + 255) / 256, 256, 0, stream>>>(Wn, wnt, 6, 128, 128);

  const int GRID = 8 * 64;  // one block per (n,h) row
  // a0 = conv(x, W0) + b0
  k_conv3x3<64,  false, false><<<GRID, 256, 0, stream>>>(xt,   nullptr, w0t,        b0,       1.f, buf0);
  // a1 = conv(a0, W1) + 1*b1                     (children of node1: [0])
  k_conv3x3<128, false, false><<<GRID, 256, 0, stream>>>(buf0, nullptr, wnt + 0*WL, bn + 0*128, 1.f, buf1);
  // a2 = conv(a0+a1, W2) + 2*b2                  (node2: [0,1])
  k_conv3x3<128, true,  false><<<GRID, 256, 0, stream>>>(buf0, buf1,    wnt + 1*WL, bn + 1*128, 2.f, buf2);
  // a3 = conv(a1+a2, W3) + 2*b3                  (node3: [1,2])
  k_conv3x3<128, true,  false><<<GRID, 256, 0, stream>>>(buf1, buf2,    wnt + 2*WL, bn + 2*128, 2.f, buf0);
  // a4 = conv(a2+a3, W4) + 2*b4                  (node4: [2,3])
  k_conv3x3<128, true,  false><<<GRID, 256, 0, stream>>>(buf2, buf0,    wnt + 3*WL, bn + 3*128, 2.f, buf1);
  // a5 = conv(a3+a4, W5) + 2*b5                  (node5: [3,4])
  k_conv3x3<128, true,  false><<<GRID, 256, 0, stream>>>(buf0, buf1,    wnt + 4*WL, bn + 4*128, 2.f, buf2);
  // a6 = conv(a4+a5, W6) + 2*b6 -> d_out (NCHW)  (node6: [4,5])
  k_conv3x3<128, true,  true ><<<GRID, 256, 0, stream>>>(buf1, buf2,    wnt + 5*WL, bn + 5*128, 2.f, out);
}